// GCNLinkPredictor_5016521801943
// MI455X (gfx1250) — compile-verified
//
#include <hip/hip_runtime.h>

typedef __attribute__((ext_vector_type(2))) float v2f;
typedef __attribute__((ext_vector_type(8))) float v8f;

#define N_C 128   // all feature widths are 128 (pair MLP input is 2*128)

// ---------------------------------------------------------------------------
// degree / normalization
// ---------------------------------------------------------------------------
__global__ void k_fill1(float* __restrict__ p, int n) {
  int i = blockIdx.x * blockDim.x + threadIdx.x;
  if (i < n) p[i] = 1.0f;   // self loop contributes 1 to every node's degree
}

__global__ void k_deg_scatter(const int* __restrict__ col, float* __restrict__ deg, int E) {
  int i = blockIdx.x * blockDim.x + threadIdx.x;
  if (i < E) atomicAdd(&deg[col[i]], 1.0f);
}

__global__ void k_rsqrt_inplace(float* __restrict__ p, int n) {
  int i = blockIdx.x * blockDim.x + threadIdx.x;
  if (i < n) {
    float d = p[i];
    p[i] = (d > 0.0f) ? rsqrtf(d) : 0.0f;
  }
}

// ---------------------------------------------------------------------------
// C[M x 128] = A[M x 128] @ B[128 x 128]   (row-major, fp32 WMMA 16x16x4)
// block = 256 threads (8 waves); block covers 32 rows x 128 cols.
// Each wave owns two 16x16 row-tiles of the same column strip, so every
// B fetch feeds two independent WMMA chains.  REQUIRES M % 32 == 0.
// ---------------------------------------------------------------------------
#define AS_LD 132   // padded LDS row stride (132 % 64 = 4 -> no bank conflicts)

__global__ void k_gemm_128(const float* __restrict__ A, const float* __restrict__ B,
                           float* __restrict__ C, int M) {
  __shared__ float As[32 * AS_LD];
  const int block_row = blockIdx.x * 32;
  const int tid  = threadIdx.x;
  const int wave = tid >> 5;
  const int lane = tid & 31;
  const int ln   = lane & 15;
  const int half = lane >> 4;     // 0: K pair {0,1}, 1: K pair {2,3} of each step
  const int n0   = wave * 16;

  // stage the 32x128 A tile into LDS with b128 loads/stores
  for (int i = tid; i < 32 * 32; i += 256) {
    int r = i >> 5, c4 = i & 31;
    float4 v = ((const float4*)A)[(long)(block_row + r) * 32 + c4];
    *(float4*)&As[r * AS_LD + c4 * 4] = v;
  }
  __syncthreads();

  v8f acc0 = {}, acc1 = {};
#pragma unroll 4
  for (int k0 = 0; k0 < N_C; k0 += 4) {
    const int ka = k0 + 2 * half;
    v2f a0, a1, b;
    a0.x = As[ln * AS_LD + ka];
    a0.y = As[ln * AS_LD + ka + 1];
    a1.x = As[(16 + ln) * AS_LD + ka];
    a1.y = As[(16 + ln) * AS_LD + ka + 1];
    b.x = B[(ka)     * N_C + n0 + ln];
    b.y = B[(ka + 1) * N_C + n0 + ln];
    acc0 = __builtin_amdgcn_wmma_f32_16x16x4_f32(false, a0, false, b, (short)0, acc0, false, false);
    acc1 = __builtin_amdgcn_wmma_f32_16x16x4_f32(false, a1, false, b, (short)0, acc1, false, false);
  }

  // D layout: VGPR r -> row (r + 8*half), col = n0 + ln   (no guards: M%32==0)
#pragma unroll
  for (int r = 0; r < 8; ++r) {
    const long row0 = block_row + r + 8 * half;
    C[row0 * N_C + n0 + ln]        = acc0[r];
    C[(row0 + 16) * N_C + n0 + ln] = acc1[r];
  }
}

// ---------------------------------------------------------------------------
// aggregation: agg = dinv^2 * T + bias  (self loop + bias folded in),
// then per-edge L2 float atomics, then optional relu.  float4 elementwise.
// ---------------------------------------------------------------------------
__global__ void k_agg_init4(const float4* __restrict__ T4, const float* __restrict__ dinv,
                            const float4* __restrict__ bias4, float4* __restrict__ agg4,
                            int total4) {
  int i = blockIdx.x * blockDim.x + threadIdx.x;
  if (i < total4) {
    int n = i >> 5;              // 32 float4 per node row
    int c4 = i & 31;
    float d = dinv[n];
    float s = d * d;
    float4 t = T4[i], b = bias4[c4];
    float4 r;
    r.x = s * t.x + b.x; r.y = s * t.y + b.y;
    r.z = s * t.z + b.z; r.w = s * t.w + b.w;
    agg4[i] = r;
  }
}

// one wave per edge: 32 lanes x float4 = 128 channels
__global__ void k_agg_scatter(const float* __restrict__ T, const float* __restrict__ dinv,
                              const int* __restrict__ row, const int* __restrict__ col,
                              float* __restrict__ agg, int E) {
  int gid = blockIdx.x * blockDim.x + threadIdx.x;
  int e = gid >> 5;
  if (e >= E) return;
  int c4 = (gid & 31) * 4;
  int r = row[e], c = col[e];
  float nrm = dinv[r] * dinv[c];
  float4 v = *(const float4*)(T + (long)r * N_C + c4);
  float* a = agg + (long)c * N_C + c4;
  atomicAdd(a + 0, nrm * v.x);
  atomicAdd(a + 1, nrm * v.y);
  atomicAdd(a + 2, nrm * v.z);
  atomicAdd(a + 3, nrm * v.w);
}

__global__ void k_relu_inplace4(float4* __restrict__ p, int total4) {
  int i = blockIdx.x * blockDim.x + threadIdx.x;
  if (i < total4) {
    float4 v = p[i];
    v.x = v.x > 0.0f ? v.x : 0.0f;
    v.y = v.y > 0.0f ? v.y : 0.0f;
    v.z = v.z > 0.0f ? v.z : 0.0f;
    v.w = v.w > 0.0f ? v.w : 0.0f;
    p[i] = v;
  }
}

// ---------------------------------------------------------------------------
// pair MLP: out[p] = relu(concat(H[src],H[dst]) @ Wm1 + bm1) @ Wm2 + bm2
// block = 256 threads, 32 pairs; 8 waves tile the 32x256x128 GEMM (two
// 16-pair row-tiles per wave sharing B fetches).  The second layer (dot with
// Wm2) is fused into the epilogue: bias+relu+scale in registers, shfl_xor
// reduce across the 16 lanes of each half, 8 partials/pair through LDS.
// ---------------------------------------------------------------------------
#define EF_LD  260   // 260 % 64 = 4 -> conflict-free column reads

__global__ void k_pair_mlp(const float* __restrict__ H,
                           const int* __restrict__ src, const int* __restrict__ dst,
                           const float* __restrict__ Wm1, const float* __restrict__ bm1,
                           const float* __restrict__ Wm2, const float* __restrict__ bm2,
                           float* __restrict__ out, int P) {
  __shared__ float ef[32 * EF_LD];    // 32 pairs x 256 features
  __shared__ float part[32][8];       // per-pair per-wave partial dot products
  const int p0  = blockIdx.x * 32;
  const int tid = threadIdx.x;

  // gather ef rows with b128 loads (H is L2-resident)
  for (int i = tid; i < 32 * 64; i += 256) {   // 64 float4 per 256-float row
    int pr = i >> 6, c4 = i & 63;
    int p = p0 + pr; if (p >= P) p = P - 1;
    int node = (c4 < 32) ? src[p] : dst[p];
    float4 v = ((const float4*)H)[(long)node * 32 + (c4 & 31)];
    *(float4*)&ef[pr * EF_LD + c4 * 4] = v;
  }
  __syncthreads();

  const int wave = tid >> 5;
  const int lane = tid & 31;
  const int ln   = lane & 15;
  const int half = lane >> 4;
  const int n0   = wave * 16;

  v8f acc0 = {}, acc1 = {};
#pragma unroll 4
  for (int k0 = 0; k0 < 2 * N_C; k0 += 4) {
    const int ka = k0 + 2 * half;
    v2f a0, a1, b;
    a0.x = ef[ln * EF_LD + ka];
    a0.y = ef[ln * EF_LD + ka + 1];
    a1.x = ef[(16 + ln) * EF_LD + ka];
    a1.y = ef[(16 + ln) * EF_LD + ka + 1];
    b.x = Wm1[(ka)     * N_C + n0 + ln];
    b.y = Wm1[(ka + 1) * N_C + n0 + ln];
    acc0 = __builtin_amdgcn_wmma_f32_16x16x4_f32(false, a0, false, b, (short)0, acc0, false, false);
    acc1 = __builtin_amdgcn_wmma_f32_16x16x4_f32(false, a1, false, b, (short)0, acc1, false, false);
  }

  // fused epilogue: relu(acc + bm1[n]) * Wm2[n], reduced over this wave's n's
  const float bn = bm1[n0 + ln];
  const float wn = Wm2[n0 + ln];
#pragma unroll
  for (int r = 0; r < 8; ++r) {
    float v0 = acc0[r] + bn; v0 = v0 > 0.0f ? v0 : 0.0f;
    float v1 = acc1[r] + bn; v1 = v1 > 0.0f ? v1 : 0.0f;
    float t0 = v0 * wn, t1 = v1 * wn;
    // sum over the 16 lanes of this half (lanes 0-15 / 16-31 stay disjoint)
    t0 += __shfl_xor(t0, 1, 32);  t1 += __shfl_xor(t1, 1, 32);
    t0 += __shfl_xor(t0, 2, 32);  t1 += __shfl_xor(t1, 2, 32);
    t0 += __shfl_xor(t0, 4, 32);  t1 += __shfl_xor(t1, 4, 32);
    t0 += __shfl_xor(t0, 8, 32);  t1 += __shfl_xor(t1, 8, 32);
    if (ln == 0) {
      part[r + 8 * half][wave]      = t0;   // pairs 0..15
      part[16 + r + 8 * half][wave] = t1;   // pairs 16..31
    }
  }
  __syncthreads();

  if (tid < 32) {
    float s = bm2[0];
#pragma unroll
    for (int w = 0; w < 8; ++w) s += part[tid][w];
    if (p0 + tid < P) out[p0 + tid] = s;
  }
}

// ---------------------------------------------------------------------------
// launch
// ---------------------------------------------------------------------------
extern "C" void kernel_launch(void* const* d_in, const int* in_sizes, int n_in,
                              void* d_out, int out_size, void* d_ws, size_t ws_size,
                              hipStream_t stream) {
  const float* x   = (const float*)d_in[0];
  const int*   ei  = (const int*)  d_in[1];
  const int*   ep  = (const int*)  d_in[2];
  const float* W1  = (const float*)d_in[3];
  const float* b1  = (const float*)d_in[4];
  const float* W2  = (const float*)d_in[5];
  const float* b2  = (const float*)d_in[6];
  const float* Wm1 = (const float*)d_in[7];
  const float* bm1 = (const float*)d_in[8];
  const float* Wm2 = (const float*)d_in[9];
  const float* bm2 = (const float*)d_in[10];
  float* out = (float*)d_out;

  const int N = in_sizes[0] / N_C;   // 100000 (multiple of 32)
  const int E = in_sizes[1] / 2;
  const int P = in_sizes[2] / 2;     // 500000 (multiple of 32)
  const int* row = ei;
  const int* col = ei + E;
  const int* src = ep;
  const int* dst = ep + P;

  // workspace: dinv [N] | bufA [N*128] | bufB [N*128]
  char* ws = (char*)d_ws;
  float* dinv = (float*)ws;
  size_t off = ((size_t)N * sizeof(float) + 255) & ~(size_t)255;
  float* bufA = (float*)(ws + off);
  off += (size_t)N * N_C * sizeof(float);
  off = (off + 255) & ~(size_t)255;
  float* bufB = (float*)(ws + off);

  const int NT = 256;
  const int total4 = N * (N_C / 4);

  // degrees -> dinv
  k_fill1<<<(N + NT - 1) / NT, NT, 0, stream>>>(dinv, N);
  k_deg_scatter<<<(E + NT - 1) / NT, NT, 0, stream>>>(col, dinv, E);
  k_rsqrt_inplace<<<(N + NT - 1) / NT, NT, 0, stream>>>(dinv, N);

  // layer 1: T1 = X @ W1 ; H1 = relu(agg(T1) + b1)
  k_gemm_128<<<N / 32, NT, 0, stream>>>(x, W1, bufA, N);
  k_agg_init4<<<(total4 + NT - 1) / NT, NT, 0, stream>>>(
      (const float4*)bufA, dinv, (const float4*)b1, (float4*)bufB, total4);
  k_agg_scatter<<<(E * 32 + NT - 1) / NT, NT, 0, stream>>>(bufA, dinv, row, col, bufB, E);
  k_relu_inplace4<<<(total4 + NT - 1) / NT, NT, 0, stream>>>((float4*)bufB, total4);

  // layer 2: T2 = H1 @ W2 ; H2 = agg(T2) + b2   (reuse bufA, then bufB)
  k_gemm_128<<<N / 32, NT, 0, stream>>>(bufB, W2, bufA, N);
  k_agg_init4<<<(total4 + NT - 1) / NT, NT, 0, stream>>>(
      (const float4*)bufA, dinv, (const float4*)b2, (float4*)bufB, total4);
  k_agg_scatter<<<(E * 32 + NT - 1) / NT, NT, 0, stream>>>(bufA, dinv, row, col, bufB, E);

  // pairwise link MLP (fused two-layer head)
  k_pair_mlp<<<P / 32, NT, 0, stream>>>(bufB, src, dst, Wm1, bm1, Wm2, bm2, out, P);
}